// TriangleOutgoing_89661737271791
// MI455X (gfx1250) — compile-verified
//
#include <hip/hip_runtime.h>
#include <cstdint>

// ---------------------------------------------------------------------------
// Triangle-outgoing block for MI455X (gfx1250), fp32 WMMA everywhere,
// async global->LDS double-buffered staging for the batched GEMM.
// B=1, CZ=C=128, H=W=384.
// ---------------------------------------------------------------------------

#define HWDIM 384
#define NCH   128
#define NPIX  (HWDIM * HWDIM)              // 147456
#define NTOT  ((size_t)NCH * (size_t)NPIX) // 18,874,368
#define EPSLN 1e-5f
#define RED_BLOCKS 4096
#define LSTR 68  // LDS row stride (floats): 272B rows -> 16B-aligned b128 dsts,
                 // conflict-free ds_load_b64 fragment reads

typedef float v2f __attribute__((ext_vector_type(2)));
typedef float v8f __attribute__((ext_vector_type(8)));

// ---- gfx1250 async global->LDS path (ASYNCcnt-tracked, bypasses VGPRs) ----
#if defined(__HIP_DEVICE_COMPILE__)
#if __has_builtin(__builtin_amdgcn_global_load_async_to_lds_b128) && \
    __has_builtin(__builtin_amdgcn_s_wait_asynccnt)
#define USE_ASYNC_COPY 1
#else
#define USE_ASYNC_COPY 0
#warning "gfx1250 async-to-LDS builtins unavailable on device; using synchronous staging"
#endif
#else
#define USE_ASYNC_COPY 0
#endif

#if USE_ASYNC_COPY
// Builtin signature (from clang diagnostic): param0 = int4* in AS(1) (global
// source), param1 = int4* in AS(3) (LDS dest), then imm offset, imm cpol.
typedef int v4i_gv __attribute__((vector_size(16)));
typedef __attribute__((address_space(1))) v4i_gv* gas_v4i_ptr;
typedef __attribute__((address_space(3))) v4i_gv* las_v4i_ptr;
#define ASYNC_COPY_B128(lds_ptr, g_ptr)                          \
  __builtin_amdgcn_global_load_async_to_lds_b128(                \
      (gas_v4i_ptr)(g_ptr), (las_v4i_ptr)(lds_ptr), 0, 0)
#endif

// V_WMMA_F32_16X16X4_F32: D(16x16,f32) = A(16x4,f32) x B(4x16,f32) + C
__device__ __forceinline__ v8f wmma_f32(v2f a, v2f b, v8f c) {
  return __builtin_amdgcn_wmma_f32_16x16x4_f32(false, a, false, b, (short)0, c,
                                               false, false);
}

__device__ __forceinline__ float sigm(float v) {
  return 1.0f / (1.0f + __expf(-v));
}

// ---------------------------------------------------------------------------
// Stage 1a: per-block partial sum / sum-of-squares (deterministic tree)
// ---------------------------------------------------------------------------
__global__ __launch_bounds__(256) void reduce_partial_kernel(
    const float4* __restrict__ src, float* __restrict__ part, int n4) {
  __shared__ float s0[256];
  __shared__ float s1[256];
  float s = 0.f, q = 0.f;
  for (int i = blockIdx.x * blockDim.x + threadIdx.x; i < n4;
       i += gridDim.x * blockDim.x) {
    float4 v = src[i];
    s += v.x + v.y + v.z + v.w;
    q += v.x * v.x + v.y * v.y + v.z * v.z + v.w * v.w;
  }
  s0[threadIdx.x] = s;
  s1[threadIdx.x] = q;
  __syncthreads();
  for (int off = 128; off > 0; off >>= 1) {
    if ((int)threadIdx.x < off) {
      s0[threadIdx.x] += s0[threadIdx.x + off];
      s1[threadIdx.x] += s1[threadIdx.x + off];
    }
    __syncthreads();
  }
  if (threadIdx.x == 0) {
    part[2 * blockIdx.x]     = s0[0];
    part[2 * blockIdx.x + 1] = s1[0];
  }
}

// ---------------------------------------------------------------------------
// Stage 1b: final reduce -> scal[0]=mean, scal[1]=rsqrt(var+eps)
// ---------------------------------------------------------------------------
__global__ __launch_bounds__(256) void reduce_final_kernel(
    const float* __restrict__ part, float* __restrict__ scal, int nblocks,
    float inv_n) {
  __shared__ float s0[256];
  __shared__ float s1[256];
  float s = 0.f, q = 0.f;
  for (int i = threadIdx.x; i < nblocks; i += 256) {
    s += part[2 * i];
    q += part[2 * i + 1];
  }
  s0[threadIdx.x] = s;
  s1[threadIdx.x] = q;
  __syncthreads();
  for (int off = 128; off > 0; off >>= 1) {
    if ((int)threadIdx.x < off) {
      s0[threadIdx.x] += s0[threadIdx.x + off];
      s1[threadIdx.x] += s1[threadIdx.x + off];
    }
    __syncthreads();
  }
  if (threadIdx.x == 0) {
    float m   = s0[0] * inv_n;
    float var = s1[0] * inv_n - m * m;
    scal[0]   = m;
    scal[1]   = rsqrtf(var + EPSLN);
  }
}

// ---------------------------------------------------------------------------
// Stage 2: apply layernorm elementwise (float4 vectorized)
// ---------------------------------------------------------------------------
__global__ __launch_bounds__(256) void ln_apply_kernel(
    const float4* __restrict__ src, const float4* __restrict__ g,
    const float4* __restrict__ b, const float* __restrict__ scal,
    float4* __restrict__ dst, int n4) {
  float m = scal[0], r = scal[1];
  for (int i = blockIdx.x * blockDim.x + threadIdx.x; i < n4;
       i += gridDim.x * blockDim.x) {
    float4 v = src[i], gg = g[i], bb = b[i];
    float4 o;
    o.x = (v.x - m) * r * gg.x + bb.x;
    o.y = (v.y - m) * r * gg.y + bb.y;
    o.z = (v.z - m) * r * gg.z + bb.z;
    o.w = (v.w - m) * r * gg.w + bb.w;
    dst[i] = o;
  }
}

// ---------------------------------------------------------------------------
// Stage 3: fused 5-way projection. One wave = one 16(out)x16(pixel) tile,
// K-loop in steps of 4 with V_WMMA_F32_16X16X4_F32, B-fragment (x) shared by
// all five projections. Applies sigmoid gating in the epilogue.
// Fragment layout assumptions:
//   A 16x4:  lane m = lane&15;  V0/V1 hold K = k0+2*(lane>>4), +1
//   B 4x16:  lane n = lane&15;  K = vgpr + 2*(lane>>4)
//   C/D:     lane n = lane&15;  M = vgpr + 8*(lane>>4)
// ---------------------------------------------------------------------------
__global__ __launch_bounds__(256) void proj5_kernel(
    const float* __restrict__ x,
    const float* __restrict__ wa1, const float* __restrict__ ba1,
    const float* __restrict__ wa2, const float* __restrict__ ba2,
    const float* __restrict__ wb1, const float* __restrict__ bb1,
    const float* __restrict__ wb2, const float* __restrict__ bb2,
    const float* __restrict__ wg,  const float* __restrict__ bg,
    float* __restrict__ aout, float* __restrict__ bout,
    float* __restrict__ gout) {
  int lane = threadIdx.x & 31;
  int wave = threadIdx.x >> 5;
  int lh = lane & 15, half = lane >> 4;
  int p    = (blockIdx.x * 8 + wave) * 16 + lh;  // pixel (N)
  int o0   = blockIdx.y * 16;                    // out-channel tile (M)
  int mrow = (o0 + lh) * NCH;

  v8f acc1 = {}, acc2 = {}, acc3 = {}, acc4 = {}, acc5 = {};
  for (int k0 = 0; k0 < NCH; k0 += 4) {
    int kA = k0 + 2 * half;
    v2f B;
    B.x = x[(size_t)kA * NPIX + p];
    B.y = x[(size_t)(kA + 1) * NPIX + p];
    v2f A1 = *(const v2f*)&wa1[mrow + kA];
    v2f A2 = *(const v2f*)&wa2[mrow + kA];
    v2f A3 = *(const v2f*)&wb1[mrow + kA];
    v2f A4 = *(const v2f*)&wb2[mrow + kA];
    v2f A5 = *(const v2f*)&wg[mrow + kA];
    acc1 = wmma_f32(A1, B, acc1);
    acc2 = wmma_f32(A2, B, acc2);
    acc3 = wmma_f32(A3, B, acc3);
    acc4 = wmma_f32(A4, B, acc4);
    acc5 = wmma_f32(A5, B, acc5);
  }

  int mo0 = o0 + 8 * half;
  for (int v = 0; v < 8; ++v) {
    int mo = mo0 + v;
    size_t idx = (size_t)mo * NPIX + p;
    float s1 = acc1[v] + ba1[mo];
    float s2 = acc2[v] + ba2[mo];
    aout[idx] = sigm(s1) * s2;
    float s3 = acc3[v] + bb1[mo];
    float s4 = acc4[v] + bb2[mo];
    bout[idx] = sigm(s3) * s4;
    gout[idx] = sigm(acc5[v] + bg[mo]);
  }
}

// ---------------------------------------------------------------------------
// Stage 4: batched AB^T: AB[c][i][j] = sum_k a[c][i][k] * b[c][j][k]
// 128 independent 384x384x384 GEMMs. Block = 8 waves computing a 64x64 tile.
// K chunked by 64 through LDS. On gfx1250 the staging is double-buffered via
// GLOBAL_LOAD_ASYNC_TO_LDS_B128 (ASYNCcnt): chunk t+1 streams into LDS while
// chunk t feeds the WMMAs. 8 async b128 loads per thread per chunk, so
// s_wait_asynccnt 8 == "chunk t fully landed" (in-order completion).
// ---------------------------------------------------------------------------
__global__ __launch_bounds__(256) void gemm_abt_kernel(
    const float* __restrict__ A, const float* __restrict__ B,
    float* __restrict__ AB) {
#if USE_ASYNC_COPY
  __shared__ float As[2][64 * LSTR];
  __shared__ float Bs[2][64 * LSTR];
#else
  __shared__ float As[1][64 * LSTR];
  __shared__ float Bs[1][64 * LSTR];
#endif
  int c  = blockIdx.y;
  int ti = blockIdx.x / 6, tj = blockIdx.x % 6;
  int i0 = ti * 64, j0 = tj * 64;
  int lane = threadIdx.x & 31;
  int wave = threadIdx.x >> 5;
  int lh = lane & 15, half = lane >> 4;
  int wi = wave & 3;   // i sub-tile (16 rows each)
  int wj = wave >> 2;  // j sub-tile (32 cols each)

  const float* ga = A + (size_t)c * NPIX + (size_t)i0 * HWDIM;
  const float* gb = B + (size_t)c * NPIX + (size_t)j0 * HWDIM;

  int sr = threadIdx.x >> 4;        // staging row 0..15
  int sc = (threadIdx.x & 15) * 4;  // staging col 0..60

  v8f acc0 = {}, acc1 = {};

#if USE_ASYNC_COPY
  // Prefetch chunk 0.
  for (int it = 0; it < 64; it += 16) {
    int r = sr + it;
    ASYNC_COPY_B128(&As[0][r * LSTR + sc], &ga[(size_t)r * HWDIM + sc]);
    ASYNC_COPY_B128(&Bs[0][r * LSTR + sc], &gb[(size_t)r * HWDIM + sc]);
  }
  for (int t = 0; t < HWDIM / 64; ++t) {
    int kk = t * 64;
    if (t + 1 < HWDIM / 64) {
      int nb = (t + 1) & 1;
      for (int it = 0; it < 64; it += 16) {
        int r = sr + it;
        ASYNC_COPY_B128(&As[nb][r * LSTR + sc],
                        &ga[(size_t)r * HWDIM + kk + 64 + sc]);
        ASYNC_COPY_B128(&Bs[nb][r * LSTR + sc],
                        &gb[(size_t)r * HWDIM + kk + 64 + sc]);
      }
      __builtin_amdgcn_s_wait_asynccnt(8);  // chunk t landed; t+1 in flight
    } else {
      __builtin_amdgcn_s_wait_asynccnt(0);
    }
    __syncthreads();  // all waves' chunk-t data visible
    const float* as  = &As[t & 1][(wi * 16 + lh) * LSTR];
    const float* bs0 = &Bs[t & 1][(wj * 32 + lh) * LSTR];
    const float* bs1 = bs0 + 16 * LSTR;
    for (int k0 = 0; k0 < 64; k0 += 4) {
      int kA = k0 + 2 * half;
      v2f af = *(const v2f*)&as[kA];
      v2f b0 = *(const v2f*)&bs0[kA];
      v2f b1 = *(const v2f*)&bs1[kA];
      acc0 = wmma_f32(af, b0, acc0);
      acc1 = wmma_f32(af, b1, acc1);
    }
    __syncthreads();  // done reading buf t&1; safe to overwrite next iter
  }
#else
  for (int kk = 0; kk < HWDIM; kk += 64) {
    __syncthreads();
    for (int it = 0; it < 64; it += 16) {
      int r = sr + it;
      *(float4*)&As[0][r * LSTR + sc] =
          *(const float4*)&ga[(size_t)r * HWDIM + kk + sc];
      *(float4*)&Bs[0][r * LSTR + sc] =
          *(const float4*)&gb[(size_t)r * HWDIM + kk + sc];
    }
    __syncthreads();
    const float* as  = &As[0][(wi * 16 + lh) * LSTR];
    const float* bs0 = &Bs[0][(wj * 32 + lh) * LSTR];
    const float* bs1 = bs0 + 16 * LSTR;
    for (int k0 = 0; k0 < 64; k0 += 4) {
      int kA = k0 + 2 * half;
      v2f af = *(const v2f*)&as[kA];
      v2f b0 = *(const v2f*)&bs0[kA];
      v2f b1 = *(const v2f*)&bs1[kA];
      acc0 = wmma_f32(af, b0, acc0);
      acc1 = wmma_f32(af, b1, acc1);
    }
  }
#endif

  float* gout = AB + (size_t)c * NPIX;
  int ib = i0 + wi * 16 + 8 * half;
  int ja = j0 + wj * 32 + lh;
  for (int v = 0; v < 8; ++v) {
    gout[(size_t)(ib + v) * HWDIM + ja]      = acc0[v];
    gout[(size_t)(ib + v) * HWDIM + ja + 16] = acc1[v];
  }
}

// ---------------------------------------------------------------------------
// Stage 5: final projection (wrep) of LN2(ab), gated by g -> d_out
// ---------------------------------------------------------------------------
__global__ __launch_bounds__(256) void proj_final_kernel(
    const float* __restrict__ xn, const float* __restrict__ w,
    const float* __restrict__ bias, const float* __restrict__ gate,
    float* __restrict__ out) {
  int lane = threadIdx.x & 31;
  int wave = threadIdx.x >> 5;
  int lh = lane & 15, half = lane >> 4;
  int p    = (blockIdx.x * 8 + wave) * 16 + lh;
  int o0   = blockIdx.y * 16;
  int mrow = (o0 + lh) * NCH;

  v8f acc = {};
  for (int k0 = 0; k0 < NCH; k0 += 4) {
    int kA = k0 + 2 * half;
    v2f B;
    B.x = xn[(size_t)kA * NPIX + p];
    B.y = xn[(size_t)(kA + 1) * NPIX + p];
    v2f Af = *(const v2f*)&w[mrow + kA];
    acc = wmma_f32(Af, B, acc);
  }

  int mo0 = o0 + 8 * half;
  for (int v = 0; v < 8; ++v) {
    int mo = mo0 + v;
    size_t idx = (size_t)mo * NPIX + p;
    out[idx] = gate[idx] * (acc[v] + bias[mo]);
  }
}

// ---------------------------------------------------------------------------
// Host launcher
// ---------------------------------------------------------------------------
extern "C" void kernel_launch(void* const* d_in, const int* in_sizes, int n_in,
                              void* d_out, int out_size, void* d_ws,
                              size_t ws_size, hipStream_t stream) {
  (void)in_sizes; (void)n_in; (void)out_size; (void)ws_size;

  const float* pair  = (const float*)d_in[0];
  const float* ln1g  = (const float*)d_in[1];
  const float* ln1b  = (const float*)d_in[2];
  const float* wa1w  = (const float*)d_in[3];
  const float* wa1b  = (const float*)d_in[4];
  const float* wa2w  = (const float*)d_in[5];
  const float* wa2b  = (const float*)d_in[6];
  const float* wb1w  = (const float*)d_in[7];
  const float* wb1b  = (const float*)d_in[8];
  const float* wb2w  = (const float*)d_in[9];
  const float* wb2b  = (const float*)d_in[10];
  const float* wgw   = (const float*)d_in[11];
  const float* wgb   = (const float*)d_in[12];
  const float* ln2g  = (const float*)d_in[13];
  const float* ln2b  = (const float*)d_in[14];
  const float* wrepw = (const float*)d_in[15];
  const float* wrepb = (const float*)d_in[16];
  float* out = (float*)d_out;
  float* ws  = (float*)d_ws;

  // Workspace layout (buffers overlaid across pipeline phases):
  float* xbuf = ws;                 // x  (later: ab)
  float* abuf = ws + NTOT;          // a
  float* bbuf = ws + 2 * NTOT;      // b  (later: xn2)
  float* gbuf = ws + 3 * NTOT;      // g
  float* part = ws + 4 * NTOT;      // 2*RED_BLOCKS partials
  float* scal = part + 2 * RED_BLOCKS;

  const int n4 = (int)(NTOT / 4);
  const float inv_n = 1.0f / (float)NTOT;
  dim3 pgrid(NPIX / 16 / 8, NCH / 16);  // 1152 x 8
  dim3 ggrid(36, NCH);                  // 6x6 tiles x 128 channels

  // LN1
  reduce_partial_kernel<<<RED_BLOCKS, 256, 0, stream>>>(
      (const float4*)pair, part, n4);
  reduce_final_kernel<<<1, 256, 0, stream>>>(part, scal, RED_BLOCKS, inv_n);
  ln_apply_kernel<<<4096, 256, 0, stream>>>(
      (const float4*)pair, (const float4*)ln1g, (const float4*)ln1b, scal,
      (float4*)xbuf, n4);

  // a, b, g projections (fused, 5 WMMA accumulators per tile)
  proj5_kernel<<<pgrid, 256, 0, stream>>>(xbuf, wa1w, wa1b, wa2w, wa2b, wb1w,
                                          wb1b, wb2w, wb2b, wgw, wgb, abuf,
                                          bbuf, gbuf);

  // ab = batched A * B^T  (written over xbuf, which is dead now)
  gemm_abt_kernel<<<ggrid, 256, 0, stream>>>(abuf, bbuf, xbuf);

  // LN2 over ab  (xn2 written over bbuf, which is dead now)
  reduce_partial_kernel<<<RED_BLOCKS, 256, 0, stream>>>(
      (const float4*)xbuf, part, n4);
  reduce_final_kernel<<<1, 256, 0, stream>>>(part, scal, RED_BLOCKS, inv_n);
  ln_apply_kernel<<<4096, 256, 0, stream>>>(
      (const float4*)xbuf, (const float4*)ln2g, (const float4*)ln2b, scal,
      (float4*)bbuf, n4);

  // out = g * (wrep @ xn2 + brep)
  proj_final_kernel<<<pgrid, 256, 0, stream>>>(bbuf, wrepw, wrepb, gbuf, out);
}